// Net_33157147525942
// MI455X (gfx1250) — compile-verified
//
#include <hip/hip_runtime.h>
#include <math.h>

#define NN 80000
#define PP 2000
#define GG 40
#define EE 640000
#define HH 128
#define BN_EPS    1e-5f
#define MIN_SCOREF 0.1f
#define TOLERF    1e-7f

typedef __attribute__((ext_vector_type(2))) float v2f;
typedef __attribute__((ext_vector_type(8))) float v8f;

// ---------------- attention GCN (2 -> 1, edge weight 1) ----------------

// deg1[n]=1 (self loop), attn[n]=0, hattn[n] = x[n,:] . W_attn
__global__ void init_attn(const float* __restrict__ x, const float* __restrict__ Wa,
                          float* __restrict__ deg1, float* __restrict__ attn,
                          float* __restrict__ hattn) {
    int n = blockIdx.x * 256 + threadIdx.x;
    if (n >= NN) return;
    deg1[n] = 1.0f;
    attn[n] = 0.0f;
    hattn[n] = x[2 * n] * Wa[0] + x[2 * n + 1] * Wa[1];
}

__global__ void deg_count_ones(const int* __restrict__ dst, float* __restrict__ deg1) {
    int e = blockIdx.x * 256 + threadIdx.x;
    if (e >= EE) return;
    atomicAdd(&deg1[dst[e]], 1.0f);
}

__global__ void attn_scatter(const int* __restrict__ src, const int* __restrict__ dst,
                             const float* __restrict__ hattn, const float* __restrict__ deg1,
                             float* __restrict__ attn) {
    int e = blockIdx.x * 256 + threadIdx.x;
    if (e >= EE) return;
    int s = src[e], d = dst[e];
    float norm = rsqrtf(deg1[s]) * rsqrtf(deg1[d]);
    atomicAdd(&attn[d], hattn[s] * norm);
}

__global__ void attn_finish(float* __restrict__ attn, const float* __restrict__ hattn,
                            const float* __restrict__ deg1, const float* __restrict__ ba) {
    int n = blockIdx.x * 256 + threadIdx.x;
    if (n >= NN) return;
    attn[n] = attn[n] + hattn[n] / deg1[n] + ba[0];
}

// ---------------- per-graph softmax (nodes contiguous per graph) ----------------

__global__ void seg_softmax_red(const float* __restrict__ attn, const float* __restrict__ wt_,
                                float* __restrict__ gmax, float* __restrict__ gsum) {
    __shared__ float red[256];
    int g = blockIdx.x, t = threadIdx.x;
    float wt = wt_[0];
    const float* a = attn + g * PP;
    float mx = -1e30f;
    for (int i = t; i < PP; i += 256) mx = fmaxf(mx, a[i] * wt);
    red[t] = mx; __syncthreads();
    for (int s = 128; s > 0; s >>= 1) { if (t < s) red[t] = fmaxf(red[t], red[t + s]); __syncthreads(); }
    float m = red[0]; __syncthreads();
    float sm = 0.0f;
    for (int i = t; i < PP; i += 256) sm += expf(a[i] * wt - m);
    red[t] = sm; __syncthreads();
    for (int s = 128; s > 0; s >>= 1) { if (t < s) red[t] += red[t + s]; __syncthreads(); }
    if (t == 0) { gmax[g] = m; gsum[g] = red[0]; }
}

// score, keep mask, deg2 init
__global__ void score_keep(const float* __restrict__ attn, const float* __restrict__ wt_,
                           const float* __restrict__ gmax, const float* __restrict__ gsum,
                           float* __restrict__ score, int* __restrict__ keep,
                           float* __restrict__ deg2) {
    int n = blockIdx.x * 256 + threadIdx.x;
    if (n >= NN) return;
    int g = n / PP;
    float sc = expf(attn[n] * wt_[0] - gmax[g]) / gsum[g];
    float smax = 1.0f / gsum[g];                    // max softmax value in this graph
    score[n] = sc;
    keep[n] = (sc > fminf(smax - TOLERF, MIN_SCOREF)) ? 1 : 0;
    deg2[n] = 1.0f;
}

__global__ void deg_count_kept(const int* __restrict__ src, const int* __restrict__ dst,
                               const int* __restrict__ keep, float* __restrict__ deg2) {
    int e = blockIdx.x * 256 + threadIdx.x;
    if (e >= EE) return;
    int s = src[e], d = dst[e];
    if (keep[s] && keep[d]) atomicAdd(&deg2[d], 1.0f);
}

// ---------------- GCN layer pieces ----------------

// layer0: HW = (masked x*score) @ W0  (2 -> 128); also zero Agg
__global__ void layer0_hw(const float* __restrict__ x, const float* __restrict__ score,
                          const int* __restrict__ keep, const float* __restrict__ W0,
                          float* __restrict__ HW, float* __restrict__ Agg) {
    int idx = blockIdx.x * 256 + threadIdx.x;   // NN*HH threads
    int n = idx >> 7, j = idx & 127;
    float sc = keep[n] ? score[n] : 0.0f;
    float h0 = x[2 * n] * sc, h1 = x[2 * n + 1] * sc;
    HW[idx] = h0 * W0[j] + h1 * W0[HH + j];
    Agg[idx] = 0.0f;
}

// Y = X @ W using v_wmma_f32_16x16x4_f32 (full-precision path; GEMM is not the
// bottleneck here, so no reason to downcast). Also zeroes Agg for the scatter.
// grid.x = NN/16 row blocks; 256 threads = 8 waves; wave w -> 16-col tile w.
__global__ void gemm_wmma_f32(const float* __restrict__ X, const float* __restrict__ W,
                              float* __restrict__ Y, float* __restrict__ Agg) {
    int row0  = blockIdx.x * 16;
    int wave  = threadIdx.x >> 5;
    int lane  = threadIdx.x & 31;
    int col0  = wave * 16;
    int mrow  = lane & 15;        // A row / B col / D col
    int khalf = lane >> 4;        // selects k pair {0,1} vs {2,3}

    const float* xrow = X + (row0 + mrow) * HH;
    v8f acc = {};
#pragma unroll
    for (int k = 0; k < HH; k += 4) {
        int ka = k + khalf * 2;
        v2f a = *(const v2f*)(xrow + ka);                 // A: row mrow, k=ka..ka+1
        v2f b;
        b.x = W[ka * HH + col0 + mrow];                   // B: k=ka,   col mrow
        b.y = W[(ka + 1) * HH + col0 + mrow];             // B: k=ka+1, col mrow
        acc = __builtin_amdgcn_wmma_f32_16x16x4_f32(
            false, a, false, b, (short)0, acc, false, false);
    }
#pragma unroll
    for (int v = 0; v < 8; ++v) {                          // D: vgpr v -> row v+8*khalf
        Y[(row0 + v + 8 * khalf) * HH + col0 + mrow] = acc[v];
    }
    for (int i = threadIdx.x; i < 16 * HH; i += 256)       // zero accumulator rows
        Agg[row0 * HH + i] = 0.0f;
}

// one wave per edge: lane reads float4 of source row, 4 atomic adds to dst row
__global__ void edge_scatter(const int* __restrict__ src, const int* __restrict__ dst,
                             const int* __restrict__ keep, const float* __restrict__ deg,
                             const float* __restrict__ HW, float* __restrict__ Agg) {
    int e = blockIdx.x * 8 + (threadIdx.x >> 5);
    if (e >= EE) return;
    int s = src[e], d = dst[e];
    if (!(keep[s] && keep[d])) return;
    float norm = rsqrtf(deg[s]) * rsqrtf(deg[d]);
    int lane = threadIdx.x & 31;
    float4 v = ((const float4*)(HW + (size_t)s * HH))[lane];
    float* ad = Agg + (size_t)d * HH + lane * 4;
    atomicAdd(ad + 0, v.x * norm);
    atomicAdd(ad + 1, v.y * norm);
    atomicAdd(ad + 2, v.z * norm);
    atomicAdd(ad + 3, v.w * norm);
}

// out = BN(relu(Agg + HW/deg + b)); Out may alias Agg (pure elementwise)
__global__ void gcn_finish(const float* __restrict__ HW, const float* __restrict__ Agg,
                           const float* __restrict__ deg, const float* __restrict__ b,
                           const float* __restrict__ g, const float* __restrict__ be,
                           const float* __restrict__ m, const float* __restrict__ v,
                           float* __restrict__ Out) {
    int idx = blockIdx.x * 256 + threadIdx.x;   // NN*HH threads
    int n = idx >> 7, j = idx & 127;
    float val = Agg[idx] + HW[idx] / deg[n] + b[j];
    val = fmaxf(val, 0.0f);
    val = (val - m[j]) * (g[j] * rsqrtf(v[j] + BN_EPS)) + be[j];
    Out[idx] = val;
}

// ---------------- pooling + head ----------------

__global__ void pool_max(const float* __restrict__ h, const int* __restrict__ keep,
                         float* __restrict__ pooled) {
    int g = blockIdx.x, j = threadIdx.x;       // 128 threads: coalesced row reads
    float mx = -1e30f;
    const float* base = h + (size_t)g * PP * HH;
    const int* kp = keep + g * PP;
    for (int i = 0; i < PP; ++i)
        if (kp[i]) mx = fmaxf(mx, base[i * HH + j]);
    pooled[g * HH + j] = mx;
}

__global__ void head(const float* __restrict__ pooled, const float* __restrict__ Wf,
                     const float* __restrict__ bf, float* __restrict__ out) {
    int g = threadIdx.x;
    if (g >= GG) return;
    float l0 = bf[0], l1 = bf[1], l2 = bf[2];
    const float* p = pooled + g * HH;
    for (int j = 0; j < HH; ++j) {
        float pj = p[j];
        l0 += pj * Wf[j * 3 + 0];
        l1 += pj * Wf[j * 3 + 1];
        l2 += pj * Wf[j * 3 + 2];
    }
    float mx = fmaxf(l0, fmaxf(l1, l2));
    float ls = logf(expf(l0 - mx) + expf(l1 - mx) + expf(l2 - mx)) + mx;
    out[g * 3 + 0] = l0 - ls;
    out[g * 3 + 1] = l1 - ls;
    out[g * 3 + 2] = l2 - ls;
}

// ---------------- launcher ----------------

extern "C" void kernel_launch(void* const* d_in, const int* in_sizes, int n_in,
                              void* d_out, int out_size, void* d_ws, size_t ws_size,
                              hipStream_t stream) {
    const float* x      = (const float*)d_in[0];
    const int*   src    = (const int*)  d_in[1];
    const int*   dst    = (const int*)  d_in[2];
    // d_in[3]=batch, d_in[4]=num_graphs: unused (nodes contiguous per graph)
    const float* W_attn = (const float*)d_in[5];
    const float* b_attn = (const float*)d_in[6];
    const float* w_topk = (const float*)d_in[7];
    const float* W0 = (const float*)d_in[8];  const float* b0 = (const float*)d_in[9];
    const float* g0 = (const float*)d_in[10]; const float* be0 = (const float*)d_in[11];
    const float* m0 = (const float*)d_in[12]; const float* v0 = (const float*)d_in[13];
    const float* W1 = (const float*)d_in[14]; const float* b1 = (const float*)d_in[15];
    const float* g1 = (const float*)d_in[16]; const float* be1 = (const float*)d_in[17];
    const float* m1 = (const float*)d_in[18]; const float* v1 = (const float*)d_in[19];
    const float* W2 = (const float*)d_in[20]; const float* b2 = (const float*)d_in[21];
    const float* g2 = (const float*)d_in[22]; const float* be2 = (const float*)d_in[23];
    const float* m2 = (const float*)d_in[24]; const float* v2 = (const float*)d_in[25];
    const float* Wf = (const float*)d_in[26]; const float* bf = (const float*)d_in[27];
    float* out = (float*)d_out;

    float* ws = (float*)d_ws;
    float* buf0 = ws; ws += (size_t)NN * HH;
    float* buf1 = ws; ws += (size_t)NN * HH;
    float* buf2 = ws; ws += (size_t)NN * HH;
    float* attn = ws; ws += NN;
    float* deg1 = ws; ws += NN;
    float* deg2 = ws; ws += NN;
    float* scor = ws; ws += NN;          // doubles as hattn before softmax
    int*   keep = (int*)ws; ws += NN;
    float* gmax = ws; ws += GG;
    float* gsum = ws; ws += GG;
    float* pooled = ws; ws += (size_t)GG * HH;

    const int NB  = (NN + 255) / 256;
    const int EB  = (EE + 255) / 256;
    const int NHB = (NN * HH) / 256;     // exact

    // attention GCN + topk mask
    init_attn<<<NB, 256, 0, stream>>>(x, W_attn, deg1, attn, scor);
    deg_count_ones<<<EB, 256, 0, stream>>>(dst, deg1);
    attn_scatter<<<EB, 256, 0, stream>>>(src, dst, scor, deg1, attn);
    attn_finish<<<NB, 256, 0, stream>>>(attn, scor, deg1, b_attn);
    seg_softmax_red<<<GG, 256, 0, stream>>>(attn, w_topk, gmax, gsum);
    score_keep<<<NB, 256, 0, stream>>>(attn, w_topk, gmax, gsum, scor, keep, deg2);
    deg_count_kept<<<EB, 256, 0, stream>>>(src, dst, keep, deg2);

    // layer 0 (2 -> 128): HW=buf0, Agg=buf1, out in place -> buf1
    layer0_hw<<<NHB, 256, 0, stream>>>(x, scor, keep, W0, buf0, buf1);
    edge_scatter<<<EE / 8, 256, 0, stream>>>(src, dst, keep, deg2, buf0, buf1);
    gcn_finish<<<NHB, 256, 0, stream>>>(buf0, buf1, deg2, b0, g0, be0, m0, v0, buf1);

    // layer 1 (128 -> 128, WMMA): X=buf1, HW=buf2, Agg=buf0, out -> buf0
    gemm_wmma_f32<<<NN / 16, 256, 0, stream>>>(buf1, W1, buf2, buf0);
    edge_scatter<<<EE / 8, 256, 0, stream>>>(src, dst, keep, deg2, buf2, buf0);
    gcn_finish<<<NHB, 256, 0, stream>>>(buf2, buf0, deg2, b1, g1, be1, m1, v1, buf0);

    // layer 2 (128 -> 128, WMMA): X=buf0, HW=buf1, Agg=buf2, out -> buf2
    gemm_wmma_f32<<<NN / 16, 256, 0, stream>>>(buf0, W2, buf1, buf2);
    edge_scatter<<<EE / 8, 256, 0, stream>>>(src, dst, keep, deg2, buf1, buf2);
    gcn_finish<<<NHB, 256, 0, stream>>>(buf1, buf2, deg2, b2, g2, be2, m2, v2, buf2);

    // pooling + classifier head
    pool_max<<<GG, HH, 0, stream>>>(buf2, keep, pooled);
    head<<<1, 64, 0, stream>>>(pooled, Wf, bf, out);
}